// WeightedCrossError_48309792145884
// MI455X (gfx1250) — compile-verified
//
#include <hip/hip_runtime.h>
#include <hip/hip_bf16.h>

typedef __attribute__((ext_vector_type(2))) float v2f;
typedef __attribute__((ext_vector_type(4))) float v4f;
typedef __attribute__((ext_vector_type(8))) float v8f;

#define N_CLASSES 8

// WEIGHT_MATRIX = _W / 10.0, row-major [true][pred]
__device__ __constant__ float WNORM[64] = {
    0.0f, 0.1f, 0.6f, 1.0f, 1.0f, 1.0f, 1.0f, 1.0f,
    0.1f, 0.0f, 0.3f, 1.0f, 1.0f, 1.0f, 1.0f, 1.0f,
    0.6f, 0.3f, 0.0f, 0.2f, 0.9f, 1.0f, 1.0f, 1.0f,
    1.0f, 1.0f, 0.2f, 0.0f, 0.9f, 0.9f, 1.0f, 1.0f,
    1.0f, 1.0f, 0.9f, 0.9f, 0.0f, 0.8f, 0.8f, 0.8f,
    1.0f, 1.0f, 1.0f, 0.9f, 0.8f, 0.0f, 0.9f, 0.8f,
    1.0f, 1.0f, 1.0f, 1.0f, 0.8f, 0.9f, 0.0f, 0.9f,
    1.0f, 1.0f, 1.0f, 1.0f, 0.8f, 0.8f, 0.9f, 0.0f};

// ---------------------------------------------------------------------------
// Kernel 0: zero the 64-bin global histogram in workspace.
// ---------------------------------------------------------------------------
__global__ void zero_hist_kernel(unsigned int* __restrict__ hist) {
  if (threadIdx.x < 64) hist[threadIdx.x] = 0u;
}

// ---------------------------------------------------------------------------
// Kernel 1: streaming pass. One row = 8 f32 logits (32 B) + one int32 target.
// ~302 MB read once (> 192 MB L2, zero reuse) -> non-temporal loads (TH=NT)
// so the stream doesn't thrash L2. argmax8 (first-max-wins, matching
// jnp.argmax), per-block LDS histogram, integer global atomics (associative
// -> bit-deterministic). Bandwidth floor ~13 us at 23.3 TB/s.
// ---------------------------------------------------------------------------
__global__ void __launch_bounds__(256)
conf_hist_kernel(const float* __restrict__ inp, const int* __restrict__ tgt,
                 unsigned int* __restrict__ hist, int n) {
  __shared__ unsigned int lhist[64];
  const int t = threadIdx.x;
  if (t < 64) lhist[t] = 0u;
  __syncthreads();

  const v4f* __restrict__ in4 = reinterpret_cast<const v4f*>(inp);
  const long long stride = (long long)gridDim.x * blockDim.x;
  for (long long r = (long long)blockIdx.x * blockDim.x + t; r < n; r += stride) {
    // Two global_load_b128 ... th:TH_LOAD_NT per row (32 B, 16 B aligned).
    const v4f lo = __builtin_nontemporal_load(&in4[2 * r]);
    const v4f hi = __builtin_nontemporal_load(&in4[2 * r + 1]);
    const float v0 = lo.x, v1 = lo.y, v2 = lo.z, v3 = lo.w;
    const float v4 = hi.x, v5 = hi.y, v6 = hi.z, v7 = hi.w;

    // First-occurrence argmax over 8 values (strict '>' keeps earliest max).
    int best = 0;
    float bv = v0;
    if (v1 > bv) { bv = v1; best = 1; }
    if (v2 > bv) { bv = v2; best = 2; }
    if (v3 > bv) { bv = v3; best = 3; }
    if (v4 > bv) { bv = v4; best = 4; }
    if (v5 > bv) { bv = v5; best = 5; }
    if (v6 > bv) { bv = v6; best = 6; }
    if (v7 > bv) { bv = v7; best = 7; }

    const int bin = __builtin_nontemporal_load(&tgt[r]) * N_CLASSES + best;
    atomicAdd(&lhist[bin], 1u);
  }

  __syncthreads();
  if (t < 64) {
    const unsigned int c = lhist[t];
    if (c) atomicAdd(&hist[t], c);  // integer: associative -> deterministic
  }
}

// ---------------------------------------------------------------------------
// Kernel 2: one wave32. Weighted reduction of the 64-bin histogram as
// trace(A x B) with a single V_WMMA_F32_16X16X4_F32:
//   A[i][k] = count[4i+k]  (16x4),  B[k][j] = w[4j+k]  (4x16)
//   => D[i][i] = sum_k count[4i+k] * w[4i+k];  trace(D) = full 64-dot.
// ISA lane layouts (CDNA5 ISA):
//   A 16x4 f32:  lanes 0-15 VGPR0/1 = K0/K1; lanes 16-31 = K2/K3 (M = lane%16)
//   B 4x16 f32:  lanes 0-15 VGPR0/1 = K0/K1 row at N=lane; lanes 16-31 = K2/K3
// -> per-lane flat indices are IDENTICAL for A and B:
//      lane<16:  4*lane, 4*lane+1 ;  lane>=16: 4*(lane-16)+2, +3
// Counts <= 2^23 are exact in f32.
// D diag: D[r][r] @ VGPR r lane r; D[r+8][r+8] @ VGPR r lane r+24.
// ---------------------------------------------------------------------------
__global__ void __launch_bounds__(32)
wmma_finalize_kernel(const unsigned int* __restrict__ hist,
                     float* __restrict__ out, float inv_n) {
  const int lane = threadIdx.x;  // 0..31, one full wave (EXEC all 1s for WMMA)
  const int m = (lane < 16) ? lane : (lane - 16);
  const int k0 = (lane < 16) ? 0 : 2;
  const int i0 = 4 * m + k0;
  const int i1 = i0 + 1;

  v2f a, b;
  a.x = (float)hist[i0];
  a.y = (float)hist[i1];
  b.x = WNORM[i0];
  b.y = WNORM[i1];

  v8f c = {};
  // 8 args: (neg_a, A, neg_b, B, c_mod, C, reuse_a, reuse_b)
  v8f d = __builtin_amdgcn_wmma_f32_16x16x4_f32(
      false, a, false, b, (short)0, c, false, false);

  // Gather the 16 diagonal elements, then wave32 shuffle reduction.
  float s = 0.0f;
#pragma unroll
  for (int r = 0; r < 8; ++r) {
    s += (lane == r) ? d[r] : 0.0f;        // D[r][r]
    s += (lane == r + 24) ? d[r] : 0.0f;   // D[r+8][r+8]
  }
#pragma unroll
  for (int off = 16; off > 0; off >>= 1) s += __shfl_xor(s, off, 32);

  if (lane == 0) out[0] = s * inv_n;
}

// ---------------------------------------------------------------------------
extern "C" void kernel_launch(void* const* d_in, const int* in_sizes, int n_in,
                              void* d_out, int out_size, void* d_ws, size_t ws_size,
                              hipStream_t stream) {
  const float* inp = (const float*)d_in[0];   // [N, 8] f32
  const int* tgt = (const int*)d_in[1];       // [N] int32
  float* out = (float*)d_out;                 // scalar f32
  unsigned int* hist = (unsigned int*)d_ws;   // 64 bins (256 B of workspace)

  const int n = in_sizes[1];  // number of rows (8,388,608)

  zero_hist_kernel<<<1, 64, 0, stream>>>(hist);

  const int threads = 256;
  const int blocks = 4096;  // grid-stride; ~8 rows (256 B traffic) per thread
  conf_hist_kernel<<<blocks, threads, 0, stream>>>(inp, tgt, hist, n);

  wmma_finalize_kernel<<<1, 32, 0, stream>>>(hist, out, 1.0f / (float)n);
}